// GCNLayer_4638564679685
// MI455X (gfx1250) — compile-verified
//
#include <hip/hip_runtime.h>
#include <hip/hip_bf16.h>

#define N_NODES 100000
#define N_EDGES 1600000
#define D 128
#define M_TILES (N_NODES / 16)   // 6250 exact

typedef __attribute__((ext_vector_type(2))) float v2f;
typedef __attribute__((ext_vector_type(4))) float v4f;
typedef __attribute__((ext_vector_type(8))) float v8f;

// ---------------------------------------------------------------------------
// Kernel 1: h = x @ W using V_WMMA_F32_16X16X4_F32 (exact fp32 matmul).
// One wave computes a 16x128 output stripe (8 accumulators of 16x16).
// blockDim = 128 (4 waves). ISA layouts (CDNA5 ISA §7.12.2):
//   A 16x4 f32:  lane&15 = M row; VGPR0 = K(0|2), VGPR1 = K(1|3) by lane>>4
//   B 4x16  f32: lane&15 = N col; VGPR0 = K(0|2), VGPR1 = K(1|3) by lane>>4
//   C/D 16x16:   VGPR j -> M = j + 8*(lane>>4), N = lane&15
// ---------------------------------------------------------------------------
__global__ void __launch_bounds__(128)
gemm_wmma_f32(const float* __restrict__ x, const float* __restrict__ W,
              float* __restrict__ h) {
    const int lane = threadIdx.x & 31;
    const int wave = threadIdx.x >> 5;
    const int tile = blockIdx.x * 4 + wave;
    if (tile >= M_TILES) return;          // wave-uniform guard (EXEC all-1s inside)

    const int m0 = tile << 4;
    const int r  = lane & 15;             // M row (A) / N col (B,C,D)
    const int kh = lane >> 4;             // K half-select

    v8f acc[8];
#pragma unroll
    for (int i = 0; i < 8; ++i)
        acc[i] = v8f{0.f, 0.f, 0.f, 0.f, 0.f, 0.f, 0.f, 0.f};

    const float* xrow = x + (size_t)(m0 + r) * D;

    for (int k = 0; k < D; k += 4) {
        v2f a;
        a.x = xrow[k + 2 * kh + 0];
        a.y = xrow[k + 2 * kh + 1];
#pragma unroll
        for (int nt = 0; nt < 8; ++nt) {
            const float* wp = W + (size_t)(k + 2 * kh) * D + nt * 16 + r;
            v2f b;
            b.x = wp[0];      // K = k + 2*kh
            b.y = wp[D];      // K = k + 2*kh + 1
            acc[nt] = __builtin_amdgcn_wmma_f32_16x16x4_f32(
                false, a, false, b, (short)0, acc[nt], false, false);
        }
    }

#pragma unroll
    for (int nt = 0; nt < 8; ++nt) {
#pragma unroll
        for (int j = 0; j < 8; ++j) {
            h[(size_t)(m0 + j + 8 * kh) * D + nt * 16 + r] = acc[nt][j];
        }
    }
}

// ---------------------------------------------------------------------------
// Kernel 2: out[n, c] = bias[c]  (zero-init + bias in one pass, b128 stores)
// 12.8M floats -> 3.2M float4 stores.
// ---------------------------------------------------------------------------
__global__ void __launch_bounds__(256)
init_out(v4f* __restrict__ out, const float* __restrict__ bias) {
    size_t i = (size_t)blockIdx.x * 256 + threadIdx.x;   // float4 index
    if (i < (size_t)N_NODES * (D / 4)) {
        int c = (int)(i & (D / 4 - 1)) * 4;
        out[i] = v4f{bias[c], bias[c + 1], bias[c + 2], bias[c + 3]};
    }
}

// ---------------------------------------------------------------------------
// Kernel 3: per-edge gather + relu + scatter-add.
// One wave32 per edge: 128 floats -> float4 per lane (b128 loads).
//  - edge_attr is streamed once -> non-temporal loads (TH=NT) so the
//    L2-resident hot data (h: 51MB gather table, out: 51MB atomic target)
//    stays cached in the 192MB L2.
//  - scatter uses relaxed agent-scope fetch_add -> native
//    global_atomic_add_f32 (non-returning).
// blockDim = 256 (8 waves / 8 edges per block).
// ---------------------------------------------------------------------------
__global__ void __launch_bounds__(256)
edge_scatter(const long long* __restrict__ ei, const float* __restrict__ ea,
             const float* __restrict__ h, float* __restrict__ out) {
    const int lane = threadIdx.x & 31;
    const int wave = threadIdx.x >> 5;
    const long long e = (long long)blockIdx.x * 8 + wave;
    if (e >= N_EDGES) return;

    const long long s = ei[e];            // src node (gather h)
    const long long d = ei[N_EDGES + e];  // dst node (scatter out)

    const v4f* hp = (const v4f*)(h + (size_t)s * D);
    const v4f* ap = (const v4f*)(ea + (size_t)e * D);
    float*     op = out + (size_t)d * D;

    v4f hv = hp[lane];                              // L2-temporal gather
    v4f av = __builtin_nontemporal_load(ap + lane); // streamed once -> NT

    float m0 = fmaxf(hv.x + av.x, 0.f);
    float m1 = fmaxf(hv.y + av.y, 0.f);
    float m2 = fmaxf(hv.z + av.z, 0.f);
    float m3 = fmaxf(hv.w + av.w, 0.f);

    const int c = lane * 4;
    __hip_atomic_fetch_add(op + c + 0, m0, __ATOMIC_RELAXED, __HIP_MEMORY_SCOPE_AGENT);
    __hip_atomic_fetch_add(op + c + 1, m1, __ATOMIC_RELAXED, __HIP_MEMORY_SCOPE_AGENT);
    __hip_atomic_fetch_add(op + c + 2, m2, __ATOMIC_RELAXED, __HIP_MEMORY_SCOPE_AGENT);
    __hip_atomic_fetch_add(op + c + 3, m3, __ATOMIC_RELAXED, __HIP_MEMORY_SCOPE_AGENT);
}

// ---------------------------------------------------------------------------
extern "C" void kernel_launch(void* const* d_in, const int* in_sizes, int n_in,
                              void* d_out, int out_size, void* d_ws, size_t ws_size,
                              hipStream_t stream) {
    const float*     x    = (const float*)d_in[0];
    const long long* ei   = (const long long*)d_in[1];   // int64 [2, E]
    const float*     ea   = (const float*)d_in[2];
    const float*     W    = (const float*)d_in[3];
    const float*     bias = (const float*)d_in[4];
    float* out = (float*)d_out;
    float* h   = (float*)d_ws;   // needs N_NODES*D*4 = 51.2 MB scratch

    // 1) h = x @ W  (WMMA fp32)
    gemm_wmma_f32<<<(M_TILES + 3) / 4, 128, 0, stream>>>(x, W, h);

    // 2) out = bias (broadcast, b128 stores)
    init_out<<<(N_NODES * (D / 4) + 255) / 256, 256, 0, stream>>>((v4f*)out, bias);

    // 3) out[dst] += relu(h[src] + edge_attr)
    edge_scatter<<<N_EDGES / 8, 256, 0, stream>>>(ei, ea, h, out);
}